// BatchDifferentiableKF_3994319586150
// MI455X (gfx1250) — compile-verified
//
#include <hip/hip_runtime.h>

// Batched Kalman filter, B=16384, T=512.
// Key structure: the gain sequence K_t is batch-independent (P0/F/Q/R shared),
// dynamics decouple per axis into affine 2-state scans -> wave-per-batch
// parallel affine scan. Memory-bound: ~192MB traffic -> ~8.3us @ 23.3TB/s.

#define KF_T   512
#define CHUNK  16      // timesteps per lane
#define WAVE   32      // gfx1250 is wave32
#define WPB    8       // waves per 256-thread block

// vector-of-4-int type matching the async-to-LDS builtin's pointer params
typedef int v4i __attribute__((vector_size(4 * sizeof(int))));
typedef __attribute__((address_space(1))) v4i* as1_v4i_ptr;
typedef __attribute__((address_space(3))) v4i* as3_v4i_ptr;

// ---------------- gain precompute (one thread, 512 steps) -----------------
// Writes gains transposed: g[step][lane] (float2 kp,kv) with t = lane*16+step,
// so the main kernel's LDS reads are conflict-free ds_load_b64.
__global__ void kf_gains_kernel(const float* __restrict__ dt_p,
                                const float* __restrict__ qpos_p,
                                const float* __restrict__ qvel_p,
                                const float* __restrict__ rvel_p,
                                float2* __restrict__ gout) {
  if (threadIdx.x != 0 || blockIdx.x != 0) return;
  const float dt   = *dt_p;
  const float qpos = *qpos_p;
  const float qvel = *qvel_p;
  const float r    = *rvel_p + 1e-6f;   // REG from reference
  // per-axis 2x2 covariance: p00=pos var, p01=cov, p11=vel var; P0 = I
  float p00 = 1.f, p01 = 0.f, p11 = 1.f;
  for (int t = 0; t < KF_T; ++t) {
    const float a00 = p00 + dt * (2.f * p01 + dt * p11) + qpos;
    const float a01 = p01 + dt * p11;
    const float a11 = p11 + qvel;
    const float S   = a11 + r;
    const float kp  = a01 / S;
    const float kv  = a11 / S;
    gout[(t & (CHUNK - 1)) * WAVE + (t >> 4)] = make_float2(kp, kv);
    p00 = a00 - kp * a01;
    p01 = a01 - kp * a11;
    p11 = a11 - kv * a11;
  }
}

// ---------------- affine map machinery ------------------------------------
// Map: p' = p + c*v + g ; v' = a*v + b   (a,c shared across axes; b,g per axis)
struct Aff { float a, c, bx, gx, by, gy; };

__device__ __forceinline__ Aff aff_compose(const Aff hi, const Aff lo) {
  // hi applied AFTER lo
  Aff r;
  r.a  = hi.a * lo.a;
  r.c  = fmaf(hi.c, lo.a, lo.c);
  r.bx = fmaf(hi.a, lo.bx, hi.bx);
  r.by = fmaf(hi.a, lo.by, hi.by);
  r.gx = lo.gx + fmaf(hi.c, lo.bx, hi.gx);
  r.gy = lo.gy + fmaf(hi.c, lo.by, hi.gy);
  return r;
}

__device__ __forceinline__ void aff_step(Aff& acc, float kp, float kv,
                                         float dt, float zx, float zy) {
  // fold step map (a_t=1-kv, c_t=dt-kp, b_t=kv*z, g_t=kp*z) on top of acc
  const float at = 1.f - kv;
  const float ct = dt - kp;
  acc.gx = acc.gx + fmaf(ct, acc.bx, kp * zx);   // uses old bx
  acc.gy = acc.gy + fmaf(ct, acc.by, kp * zy);
  acc.bx = fmaf(at, acc.bx, kv * zx);
  acc.by = fmaf(at, acc.by, kv * zy);
  acc.c  = fmaf(ct, acc.a, acc.c);               // uses old a
  acc.a  = at * acc.a;
}

// ---------------- main scan: one wave per batch ---------------------------
__global__ __launch_bounds__(256) void kf_scan_kernel(
    const float* __restrict__ pred_vel,   // (B,T,2)
    const float* __restrict__ dt_p,
    const float* __restrict__ p0,         // (B,2)
    const float* __restrict__ gains,      // transposed table in d_ws (4KB)
    float* __restrict__ out_pos,          // (B,T,2)
    float* __restrict__ out_vel,          // (B,T,2)
    int Btotal) {
  __shared__ float2 sg[KF_T];             // 4KB gain table, [step][lane]

  const int tid = threadIdx.x;

  // Stage gain table via gfx1250 async-to-LDS (256 lanes x 16B = 4KB).
#if __has_builtin(__builtin_amdgcn_global_load_async_to_lds_b128)
  {
    char* gsrc = (char*)const_cast<float*>(gains);
    __builtin_amdgcn_global_load_async_to_lds_b128(
        (as1_v4i_ptr)(gsrc + tid * 16),
        (as3_v4i_ptr)((char*)sg + tid * 16),
        0, 0);
  }
#else
  ((float4*)sg)[tid] = ((const float4*)gains)[tid];
#endif

  const int lane = tid & (WAVE - 1);
  const int wave = tid >> 5;
  const int b = blockIdx.x * WPB + wave;
  if (b >= Btotal) return;                // whole-wave uniform exit

  const float dtv = *dt_p;

  // Lane's measurement chunk: t in [lane*16, lane*16+16), 128B contiguous.
  const float4* zp = (const float4*)(pred_vel + (size_t)b * (KF_T * 2)) + lane * 8;
  float4 z4[8];
#pragma unroll
  for (int i = 0; i < 8; ++i) z4[i] = zp[i];   // 8 independent b128 loads

#if __has_builtin(__builtin_amdgcn_global_load_async_to_lds_b128)
 #if __has_builtin(__builtin_amdgcn_s_wait_asynccnt)
  __builtin_amdgcn_s_wait_asynccnt(0);
 #else
  asm volatile("s_wait_asynccnt 0" ::: "memory");
 #endif
#endif
  __syncthreads();

  // ---- pass 1: compose this lane's 16-step affine map --------------------
  Aff m; m.a = 1.f; m.c = 0.f; m.bx = 0.f; m.gx = 0.f; m.by = 0.f; m.gy = 0.f;
#pragma unroll
  for (int i = 0; i < 8; ++i) {
    const float2 g0 = sg[(2 * i) * WAVE + lane];       // conflict-free b64
    const float2 g1 = sg[(2 * i + 1) * WAVE + lane];
    aff_step(m, g0.x, g0.y, dtv, z4[i].x, z4[i].y);
    aff_step(m, g1.x, g1.y, dtv, z4[i].z, z4[i].w);
  }

  // ---- wave32 inclusive scan (Hillis-Steele) over map composition --------
#pragma unroll
  for (int d = 1; d < WAVE; d <<= 1) {
    Aff o;
    o.a  = __shfl_up(m.a,  (unsigned)d, WAVE);
    o.c  = __shfl_up(m.c,  (unsigned)d, WAVE);
    o.bx = __shfl_up(m.bx, (unsigned)d, WAVE);
    o.gx = __shfl_up(m.gx, (unsigned)d, WAVE);
    o.by = __shfl_up(m.by, (unsigned)d, WAVE);
    o.gy = __shfl_up(m.gy, (unsigned)d, WAVE);
    const Aff cmb = aff_compose(m, o);                 // m after o
    const bool use = (lane >= d);
    m.a  = use ? cmb.a  : m.a;
    m.c  = use ? cmb.c  : m.c;
    m.bx = use ? cmb.bx : m.bx;
    m.gx = use ? cmb.gx : m.gx;
    m.by = use ? cmb.by : m.by;
    m.gy = use ? cmb.gy : m.gy;
  }
  // exclusive prefix = inclusive from lane-1; lane 0 = identity
  Aff e;
  e.bx = __shfl_up(m.bx, 1u, WAVE);
  e.gx = __shfl_up(m.gx, 1u, WAVE);
  e.by = __shfl_up(m.by, 1u, WAVE);
  e.gy = __shfl_up(m.gy, 1u, WAVE);
  if (lane == 0) { e.bx = 0.f; e.gx = 0.f; e.by = 0.f; e.gy = 0.f; }

  // entry state of this lane's chunk (v0 = 0, p0 from input)
  float vx = e.bx;
  float vy = e.by;
  float px = p0[2 * b]     + e.gx;
  float py = p0[2 * b + 1] + e.gy;

  // ---- pass 2: replay 16 steps, emit pos/vel as b128 stores --------------
  float4* op4 = (float4*)(out_pos + (size_t)b * (KF_T * 2)) + lane * 8;
  float4* ov4 = (float4*)(out_vel + (size_t)b * (KF_T * 2)) + lane * 8;
#pragma unroll
  for (int i = 0; i < 8; ++i) {
    const float2 g0 = sg[(2 * i) * WAVE + lane];
    const float2 g1 = sg[(2 * i + 1) * WAVE + lane];
    float4 pout, vout;
    // step 2i
    float yx = z4[i].x - vx, yy = z4[i].y - vy;
    px += fmaf(dtv, vx, g0.x * yx);
    py += fmaf(dtv, vy, g0.x * yy);
    vx = fmaf(g0.y, yx, vx);
    vy = fmaf(g0.y, yy, vy);
    pout.x = px; pout.y = py; vout.x = vx; vout.y = vy;
    // step 2i+1
    yx = z4[i].z - vx; yy = z4[i].w - vy;
    px += fmaf(dtv, vx, g1.x * yx);
    py += fmaf(dtv, vy, g1.x * yy);
    vx = fmaf(g1.y, yx, vx);
    vy = fmaf(g1.y, yy, vy);
    pout.z = px; pout.w = py; vout.z = vx; vout.w = vy;
    op4[i] = pout;
    ov4[i] = vout;
  }
}

// ---------------- host-side launch ----------------------------------------
extern "C" void kernel_launch(void* const* d_in, const int* in_sizes, int n_in,
                              void* d_out, int out_size, void* d_ws, size_t ws_size,
                              hipStream_t stream) {
  const float* pred_vel = (const float*)d_in[0];   // (B,T,2) f32
  const float* dt       = (const float*)d_in[1];   // scalar
  const float* p0       = (const float*)d_in[2];   // (B,2)
  const float* q_pos    = (const float*)d_in[3];   // scalar
  const float* q_vel    = (const float*)d_in[4];   // scalar
  const float* r_vel    = (const float*)d_in[5];   // scalar

  const int B = in_sizes[2] / 2;                   // 16384
  float* gains = (float*)d_ws;                     // 4KB: float2[512] transposed
  float* out_pos = (float*)d_out;
  float* out_vel = out_pos + (size_t)B * KF_T * 2;

  kf_gains_kernel<<<1, 32, 0, stream>>>(dt, q_pos, q_vel, r_vel, (float2*)gains);

  const int blocks = (B + WPB - 1) / WPB;          // one wave per batch
  kf_scan_kernel<<<blocks, 256, 0, stream>>>(pred_vel, dt, p0, gains,
                                             out_pos, out_vel, B);
}